// AFNO_70703751626992
// MI455X (gfx1250) — compile-verified
//
#include <hip/hip_runtime.h>
#include <hip/hip_bf16.h>

typedef float v2f __attribute__((ext_vector_type(2)));
typedef float v8f __attribute__((ext_vector_type(8)));

// D = A(16x4, f32) * B(4x16, f32) + C(16x16, f32)  -> v_wmma_f32_16x16x4_f32
#define WMMA_F32(a, b, c) \
  __builtin_amdgcn_wmma_f32_16x16x4_f32(false, (a), false, (b), (short)0, (c), false, false)

#define TWOPI_64 0.09817477042468103f  // 2*pi/64

// ---------------------------------------------------------------------------
// Tensor Data Mover: 2D strided tile (global -> LDS), D# per ISA 08 section 8.
// group0: count=1 | lds_addr | global_addr(57b) | type=2
// group1: data_size=4B, tensor_dim0=rowElems, tensor_dim1=tensorRows,
//         tile_dim0=rowElems, tile_dim1=tileRows, tensor_dim0_stride=rowStride
// groups 2/3 zero (2D tensor). Rows beyond tensor_dim1 zero-fill LDS (OOB).
// ---------------------------------------------------------------------------
#if __has_builtin(__builtin_amdgcn_tensor_load_to_lds)
#define HAVE_TDM 1
typedef unsigned int u32x4 __attribute__((ext_vector_type(4)));
typedef int i32x4 __attribute__((ext_vector_type(4)));
typedef int i32x8 __attribute__((ext_vector_type(8)));

__device__ __forceinline__ void tdm_load_2d(const float* gsrc, const float* ldst,
                                            unsigned rowElems, unsigned tensorRows,
                                            unsigned tileRows,
                                            unsigned long long rowStrideElems) {
  unsigned long long ga = (unsigned long long)(uintptr_t)gsrc;
  unsigned la = (unsigned)(uintptr_t)ldst;   // low 32 bits of generic ptr == LDS offset
  u32x4 g0;
  g0.x = 1u;                                   // count=1, user descriptor
  g0.y = la;                                   // lds_addr
  g0.z = (unsigned)ga;                         // global_addr[31:0]
  g0.w = (unsigned)(ga >> 32) | 0x80000000u;   // global_addr[56:32] | type=2
  i32x8 g1;
  g1[0] = 0x00020000;                          // data_size = 4 bytes
  g1[1] = (int)(rowElems << 16);               // tensor_dim0[15:0] @ bits[63:48]
  g1[2] = (int)(tensorRows << 16);             // tensor_dim1[15:0] @ bits[95:80]
  g1[3] = (int)(rowElems << 16);               // tile_dim0 @ bits[127:112]
  g1[4] = (int)tileRows;                       // tile_dim1 @ bits[143:128]
  g1[5] = (int)(unsigned)(rowStrideElems & 0xFFFFFFFFull);  // dim0_stride lo
  g1[6] = (int)(unsigned)(rowStrideElems >> 32);            // dim0_stride hi
  g1[7] = 0;
  i32x4 z4 = {0, 0, 0, 0};
#if __has_include(<hip/amd_detail/amd_gfx1250_TDM.h>)
  i32x8 z8 = {0, 0, 0, 0, 0, 0, 0, 0};
  __builtin_amdgcn_tensor_load_to_lds(g0, g1, z4, z4, z8, 0);   // 6-arg toolchain
#else
  __builtin_amdgcn_tensor_load_to_lds(g0, g1, z4, z4, 0);       // 5-arg ROCm 7.2
#endif
}

__device__ __forceinline__ void tdm_wait() {
#if __has_builtin(__builtin_amdgcn_s_wait_tensorcnt)
  __builtin_amdgcn_s_wait_tensorcnt(0);
#else
  asm volatile("s_wait_tensorcnt 0x0" ::: "memory");
#endif
  asm volatile("" ::: "memory");   // keep LDS reads below the wait
}
#endif  // HAVE_TDM

// ---- WMMA fragment movement (ISA 7.12.2 layouts, wave32) -------------------
// A (16x4): lanes 0-15 -> K={k0,k0+1}, lanes 16-31 -> K={k0+2,k0+3}; M = lane%16
__device__ __forceinline__ v2f ldsA(const float* p, int stride, int row0, int k0, int lane) {
  int m  = lane & 15;
  int kk = k0 + ((lane >> 4) << 1);
  const float* q = p + (row0 + m) * stride + kk;
  v2f a; a.x = q[0]; a.y = q[1];
  return a;
}
// B (4x16): v0: lanes0-15 K=k0, lanes16-31 K=k0+2 ; v1: K=k0+1 / k0+3 ; N = lane%16
__device__ __forceinline__ v2f ldsB(const float* p, int stride, int k0, int lane) {
  int n = lane & 15;
  int r = k0 + ((lane >> 4) << 1);
  v2f b; b.x = p[r * stride + n]; b.y = p[(r + 1) * stride + n];
  return b;
}
// C/D (16x16): VGPR j -> row = row0 + j + 8*(lane>=16); N = lane%16
__device__ __forceinline__ void stD_glb(float* p, long long rs, int row0, v8f d, int lane) {
  int n  = lane & 15;
  int rb = row0 + ((lane >> 4) << 3);
#pragma unroll
  for (int j = 0; j < 8; ++j) p[(long long)(rb + j) * rs + n] = d[j];
}
__device__ __forceinline__ void stD_lds(float* p, int stride, int row0, v8f d, int lane) {
  int n  = lane & 15;
  int rb = row0 + ((lane >> 4) << 3);
#pragma unroll
  for (int j = 0; j < 8; ++j) p[(rb + j) * stride + n] = d[j];
}

// ============================================================================
// Pass A: real DFT along W (ortho 1/8). x (2,64,64,64,128) -> R,I (...,33,128)
// one wave = one (b,d,h) line x one 16-channel tile. waves = 65536.
// ============================================================================
__global__ void __launch_bounds__(256) afno_rfft_w(const float* __restrict__ x,
                                                   float* __restrict__ R,
                                                   float* __restrict__ I) {
  extern __shared__ float smem[];
  float* Cf  = smem;              // 48*66
  float* Sf  = Cf + 48 * 66;      // 48*66
  float* stg = Sf + 48 * 66;      // 8 waves * 1024
  int wv   = threadIdx.x >> 5;
  int lane = threadIdx.x & 31;
  float* Sx = stg + wv * 1024;    // 64 x 16 staged input line

  int wid  = blockIdx.x * 8 + wv;
  int ct   = wid & 7;
  int line = wid >> 3;
  int h = line & 63, dd = (line >> 6) & 63, b = line >> 12;
  const float* xp = x + ((long long)(b * 4096 + dd * 64 + h) * 64) * 128 + ct * 16;

#ifdef HAVE_TDM
  tdm_load_2d(xp, Sx, 16, 64, 64, 128);         // DMA tile while we build twiddles
#else
  for (int i = lane; i < 1024; i += 32) Sx[i] = xp[(long long)(i >> 4) * 128 + (i & 15)];
#endif

  for (int idx = threadIdx.x; idx < 48 * 64; idx += 256) {
    int k = idx >> 6, n = idx & 63;
    float s, c;
    __sincosf(TWOPI_64 * (float)((k * n) & 63), &s, &c);
    bool v = (k < 33);
    Cf[k * 66 + n] = v ? 0.125f * c : 0.0f;
    Sf[k * 66 + n] = v ? -0.125f * s : 0.0f;
  }
  __syncthreads();
#ifdef HAVE_TDM
  tdm_wait();
#endif

  v8f cr[3] = {}, ci[3] = {};
  for (int nc = 0; nc < 16; ++nc) {
    v2f bx = ldsB(Sx, 16, nc * 4, lane);
#pragma unroll
    for (int kt = 0; kt < 3; ++kt) {
      v2f ar = ldsA(Cf, 66, kt * 16, nc * 4, lane);
      v2f ai = ldsA(Sf, 66, kt * 16, nc * 4, lane);
      cr[kt] = WMMA_F32(ar, bx, cr[kt]);
      ci[kt] = WMMA_F32(ai, bx, ci[kt]);
    }
  }
  long long ob = (long long)(b * 4096 + dd * 64 + h) * 33 * 128 + ct * 16;
  int n = lane & 15;
#pragma unroll
  for (int kt = 0; kt < 3; ++kt) {
    int rb = kt * 16 + ((lane >> 4) << 3);
#pragma unroll
    for (int j = 0; j < 8; ++j) {
      int row = rb + j;
      if (row < 33) {
        R[ob + (long long)row * 128 + n] = cr[kt][j];
        I[ob + (long long)row * 128 + n] = ci[kt][j];
      }
    }
  }
}

// ============================================================================
// Pass B/D: complex unitary DFT along H, in place. sgn=+1 fwd, -1 inv. 33792 waves.
// ============================================================================
__global__ void __launch_bounds__(256) afno_cdft_h(float* __restrict__ R,
                                                   float* __restrict__ I, float sgn) {
  extern __shared__ float smem[];
  float* Fr  = smem;              // 64*66
  float* Fi  = Fr + 64 * 66;
  float* stg = Fi + 64 * 66;      // 8 waves * 2048
  int wv   = threadIdx.x >> 5;
  int lane = threadIdx.x & 31;
  float* Sr = stg + wv * 2048;    // 64 x 16
  float* Si = Sr + 1024;

  int wid  = blockIdx.x * 8 + wv;
  int ct   = wid & 7;
  int rest = wid >> 3;
  int kw   = rest % 33;
  int tmp  = rest / 33;
  int dd = tmp & 63, b = tmp >> 6;

  long long base = ((long long)(b * 64 + dd) * 64 * 33 + kw) * 128 + ct * 16;
  const long long hs = 33LL * 128;
  float* Rp = R + base;
  float* Ip = I + base;

#ifdef HAVE_TDM
  tdm_load_2d(Rp, Sr, 16, 64, 64, (unsigned long long)hs);
  tdm_load_2d(Ip, Si, 16, 64, 64, (unsigned long long)hs);
#else
  for (int i = lane; i < 1024; i += 32) {
    int r = i >> 4, c = i & 15;
    Sr[i] = Rp[(long long)r * hs + c];
    Si[i] = Ip[(long long)r * hs + c];
  }
#endif

  for (int idx = threadIdx.x; idx < 64 * 64; idx += 256) {
    int k = idx >> 6, n = idx & 63;
    float s, c;
    __sincosf(TWOPI_64 * (float)((k * n) & 63), &s, &c);
    Fr[k * 66 + n] = 0.125f * c;
    Fi[k * 66 + n] = -0.125f * sgn * s;
  }
  __syncthreads();
#ifdef HAVE_TDM
  tdm_wait();
#endif

  v8f cr[4] = {}, ci[4] = {};
  for (int nc = 0; nc < 16; ++nc) {
    v2f br = ldsB(Sr, 16, nc * 4, lane);
    v2f bi = ldsB(Si, 16, nc * 4, lane);
    v2f biN; biN.x = -bi.x; biN.y = -bi.y;
#pragma unroll
    for (int kt = 0; kt < 4; ++kt) {
      v2f ar = ldsA(Fr, 66, kt * 16, nc * 4, lane);
      v2f ai = ldsA(Fi, 66, kt * 16, nc * 4, lane);
      cr[kt] = WMMA_F32(ar, br,  cr[kt]);   // Yr += Fr*Xr
      cr[kt] = WMMA_F32(ai, biN, cr[kt]);   // Yr -= Fi*Xi
      ci[kt] = WMMA_F32(ar, bi,  ci[kt]);   // Yi += Fr*Xi
      ci[kt] = WMMA_F32(ai, br,  ci[kt]);   // Yi += Fi*Xr
    }
  }
#pragma unroll
  for (int kt = 0; kt < 4; ++kt) {
    stD_glb(Rp, hs, kt * 16, cr[kt], lane);
    stD_glb(Ip, hs, kt * 16, ci[kt], lane);
  }
}

// ============================================================================
// Pass C: DFT along D + block complex MLP (relu, softshrink) + inverse DFT along D.
// one wave = one (b,kh,kw) line x one block (==16 channels). 33792 waves.
// ============================================================================
__global__ void __launch_bounds__(256) afno_d_mlp(float* __restrict__ R, float* __restrict__ I,
                                                  const float* __restrict__ w1,
                                                  const float* __restrict__ b1,
                                                  const float* __restrict__ w2,
                                                  const float* __restrict__ b2) {
  extern __shared__ float smem[];
  float* Fr  = smem;                 // 64*66
  float* Fi  = Fr + 64 * 66;
  float* W1r = Fi + 64 * 66;         // 8*16*16 each
  float* W1i = W1r + 2048;
  float* W2r = W1i + 2048;
  float* W2i = W2r + 2048;
  float* B1r = W2i + 2048;           // 128 each
  float* B1i = B1r + 128;
  float* B2r = B1i + 128;
  float* B2i = B2r + 128;
  float* scr = B2i + 128;            // 8 waves * 4928

  int wv   = threadIdx.x >> 5;
  int lane = threadIdx.x & 31;
  float* Sr  = scr + wv * 4928;      // 64 x 16 staged input (real)
  float* Si  = Sr + 1024;            // 64 x 16 staged input (imag)
  float* Yr  = Si + 1024;            // 64 x 18 spectrum
  float* Yi  = Yr + 1152;
  float* O1r = Yi + 1152;            // 16 x 18 layer-1 activations
  float* O1i = O1r + 288;

  int wid  = blockIdx.x * 8 + wv;
  int blk  = wid & 7;
  int rest = wid >> 3;
  int kw   = rest % 33;
  int tmp  = rest / 33;
  int kh = tmp & 63, b = tmp >> 6;

  long long base = ((long long)(b * 4096 + kh) * 33 + kw) * 128 + blk * 16;
  const long long ds = 64LL * 33 * 128;  // d stride
  float* Rp = R + base;
  float* Ip = I + base;

#ifdef HAVE_TDM
  tdm_load_2d(Rp, Sr, 16, 64, 64, (unsigned long long)ds);
  tdm_load_2d(Ip, Si, 16, 64, 64, (unsigned long long)ds);
#else
  for (int i = lane; i < 1024; i += 32) {
    int r = i >> 4, c = i & 15;
    Sr[i] = Rp[(long long)r * ds + c];
    Si[i] = Ip[(long long)r * ds + c];
  }
#endif

  for (int idx = threadIdx.x; idx < 64 * 64; idx += 256) {
    int k = idx >> 6, n = idx & 63;
    float s, c;
    __sincosf(TWOPI_64 * (float)((k * n) & 63), &s, &c);
    Fr[k * 66 + n] = 0.125f * c;
    Fi[k * 66 + n] = -0.125f * s;    // forward twiddles; inverse = conj at use site
  }
  for (int i = threadIdx.x; i < 2048; i += 256) {
    W1r[i] = w1[i]; W1i[i] = w1[2048 + i];
    W2r[i] = w2[i]; W2i[i] = w2[2048 + i];
  }
  for (int i = threadIdx.x; i < 128; i += 256) {
    B1r[i] = b1[i]; B1i[i] = b1[128 + i];
    B2r[i] = b2[i]; B2i[i] = b2[128 + i];
  }
  __syncthreads();
#ifdef HAVE_TDM
  tdm_wait();
#endif

  // ---- forward DFT along D (LDS B-frags -> LDS spectrum) ----
  {
    v8f cr[4] = {}, ci[4] = {};
    for (int nc = 0; nc < 16; ++nc) {
      v2f br = ldsB(Sr, 16, nc * 4, lane);
      v2f bi = ldsB(Si, 16, nc * 4, lane);
      v2f biN; biN.x = -bi.x; biN.y = -bi.y;
#pragma unroll
      for (int kt = 0; kt < 4; ++kt) {
        v2f ar = ldsA(Fr, 66, kt * 16, nc * 4, lane);
        v2f ai = ldsA(Fi, 66, kt * 16, nc * 4, lane);
        cr[kt] = WMMA_F32(ar, br,  cr[kt]);
        cr[kt] = WMMA_F32(ai, biN, cr[kt]);
        ci[kt] = WMMA_F32(ar, bi,  ci[kt]);
        ci[kt] = WMMA_F32(ai, br,  ci[kt]);
      }
    }
#pragma unroll
    for (int kt = 0; kt < 4; ++kt) {
      stD_lds(Yr, 18, kt * 16, cr[kt], lane);
      stD_lds(Yi, 18, kt * 16, ci[kt], lane);
    }
  }

  // ---- per-frequency block complex MLP: 64 rows x (16->16->16), 4 row tiles ----
  const float* W1rB = W1r + blk * 256;
  const float* W1iB = W1i + blk * 256;
  const float* W2rB = W2r + blk * 256;
  const float* W2iB = W2i + blk * 256;
  float b1rv = B1r[blk * 16 + (lane & 15)];
  float b1iv = B1i[blk * 16 + (lane & 15)];
  float b2rv = B2r[blk * 16 + (lane & 15)];
  float b2iv = B2i[blk * 16 + (lane & 15)];

  for (int mt = 0; mt < 4; ++mt) {
    v8f a1r = {}, a1i = {};
#pragma unroll
    for (int kc = 0; kc < 4; ++kc) {
      v2f xr = ldsA(Yr, 18, mt * 16, kc * 4, lane);
      v2f xi = ldsA(Yi, 18, mt * 16, kc * 4, lane);
      v2f xiN; xiN.x = -xi.x; xiN.y = -xi.y;
      v2f wr = ldsB(W1rB, 16, kc * 4, lane);
      v2f wi = ldsB(W1iB, 16, kc * 4, lane);
      a1r = WMMA_F32(xr,  wr, a1r);
      a1r = WMMA_F32(xiN, wi, a1r);
      a1i = WMMA_F32(xi,  wr, a1i);
      a1i = WMMA_F32(xr,  wi, a1i);
    }
    v8f t1r, t1i;
#pragma unroll
    for (int j = 0; j < 8; ++j) {
      t1r[j] = fmaxf(a1r[j] + b1rv, 0.0f);
      t1i[j] = fmaxf(a1i[j] + b1iv, 0.0f);
    }
    stD_lds(O1r, 18, 0, t1r, lane);
    stD_lds(O1i, 18, 0, t1i, lane);

    v8f a2r = {}, a2i = {};
#pragma unroll
    for (int kc = 0; kc < 4; ++kc) {
      v2f xr = ldsA(O1r, 18, 0, kc * 4, lane);
      v2f xi = ldsA(O1i, 18, 0, kc * 4, lane);
      v2f xiN; xiN.x = -xi.x; xiN.y = -xi.y;
      v2f wr = ldsB(W2rB, 16, kc * 4, lane);
      v2f wi = ldsB(W2iB, 16, kc * 4, lane);
      a2r = WMMA_F32(xr,  wr, a2r);
      a2r = WMMA_F32(xiN, wi, a2r);
      a2i = WMMA_F32(xi,  wr, a2i);
      a2i = WMMA_F32(xr,  wi, a2i);
    }
    v8f sr, si;
#pragma unroll
    for (int j = 0; j < 8; ++j) {
      float vr = a2r[j] + b2rv;
      float vi = a2i[j] + b2iv;
      sr[j] = (vr > 0.01f) ? vr - 0.01f : ((vr < -0.01f) ? vr + 0.01f : 0.0f);
      si[j] = (vi > 0.01f) ? vi - 0.01f : ((vi < -0.01f) ? vi + 0.01f : 0.0f);
    }
    stD_lds(Yr, 18, mt * 16, sr, lane);
    stD_lds(Yi, 18, mt * 16, si, lane);
  }

  // ---- inverse DFT along D (conjugate twiddles), write back to global ----
  {
    v8f cr[4] = {}, ci[4] = {};
    for (int nc = 0; nc < 16; ++nc) {
      v2f br = ldsB(Yr, 18, nc * 4, lane);
      v2f bi = ldsB(Yi, 18, nc * 4, lane);
      v2f brN; brN.x = -br.x; brN.y = -br.y;
#pragma unroll
      for (int kt = 0; kt < 4; ++kt) {
        v2f ar = ldsA(Fr, 66, kt * 16, nc * 4, lane);
        v2f ai = ldsA(Fi, 66, kt * 16, nc * 4, lane);
        cr[kt] = WMMA_F32(ar, br,  cr[kt]);  // Fr*Xr
        cr[kt] = WMMA_F32(ai, bi,  cr[kt]);  // +Fi_fwd*Xi  (conj)
        ci[kt] = WMMA_F32(ar, bi,  ci[kt]);  // Fr*Xi
        ci[kt] = WMMA_F32(ai, brN, ci[kt]);  // -Fi_fwd*Xr
      }
    }
#pragma unroll
    for (int kt = 0; kt < 4; ++kt) {
      stD_glb(Rp, ds, kt * 16, cr[kt], lane);
      stD_glb(Ip, ds, kt * 16, ci[kt], lane);
    }
  }
}

// ============================================================================
// Pass E: inverse real DFT along W with Hermitian fold + residual. 65536 waves.
// TDM declares tensor_dim1=33, tile_dim1=48: rows 33..47 zero-filled by OOB.
// ============================================================================
__global__ void __launch_bounds__(256) afno_irfft_w(const float* __restrict__ x,
                                                    const float* __restrict__ R,
                                                    const float* __restrict__ I,
                                                    float* __restrict__ out) {
  extern __shared__ float smem[];
  float* Ar_ = smem;              // 64*50
  float* Ai_ = Ar_ + 64 * 50;
  float* stg = Ai_ + 64 * 50;     // 8 waves * 1536
  int wv   = threadIdx.x >> 5;
  int lane = threadIdx.x & 31;
  float* Sr = stg + wv * 1536;    // 48 x 16
  float* Si = Sr + 768;

  int wid  = blockIdx.x * 8 + wv;
  int ct   = wid & 7;
  int line = wid >> 3;
  int h = line & 63, dd = (line >> 6) & 63, b = line >> 12;

  long long ib = (long long)(b * 4096 + dd * 64 + h) * 33 * 128 + ct * 16;
  const float* Rp = R + ib;
  const float* Ip = I + ib;

#ifdef HAVE_TDM
  tdm_load_2d(Rp, Sr, 16, 33, 48, 128);   // OOB rows (33..47) zero-filled
  tdm_load_2d(Ip, Si, 16, 33, 48, 128);
#else
  for (int i = lane; i < 768; i += 32) {
    int r = i >> 4, c = i & 15;
    Sr[i] = (r < 33) ? Rp[(long long)r * 128 + c] : 0.0f;
    Si[i] = (r < 33) ? Ip[(long long)r * 128 + c] : 0.0f;
  }
#endif

  for (int idx = threadIdx.x; idx < 64 * 48; idx += 256) {
    int w = idx / 48, k = idx - w * 48;
    float s, c;
    __sincosf(TWOPI_64 * (float)((w * k) & 63), &s, &c);
    float ck = (k == 0 || k == 32) ? 0.125f : ((k < 33) ? 0.25f : 0.0f);
    Ar_[w * 50 + k] = ck * c;
    Ai_[w * 50 + k] = -ck * s;
  }
  __syncthreads();
#ifdef HAVE_TDM
  tdm_wait();
#endif

  v8f acc[4] = {};
  for (int nc = 0; nc < 12; ++nc) {
    v2f br = ldsB(Sr, 16, nc * 4, lane);
    v2f bi = ldsB(Si, 16, nc * 4, lane);
#pragma unroll
    for (int kt = 0; kt < 4; ++kt) {
      v2f ar = ldsA(Ar_, 50, kt * 16, nc * 4, lane);
      v2f ai = ldsA(Ai_, 50, kt * 16, nc * 4, lane);
      acc[kt] = WMMA_F32(ar, br, acc[kt]);
      acc[kt] = WMMA_F32(ai, bi, acc[kt]);
    }
  }
  long long xb = (long long)(b * 4096 + dd * 64 + h) * 64 * 128 + ct * 16;
  const float* xp = x + xb;
  float* op = out + xb;
  int n = lane & 15;
#pragma unroll
  for (int kt = 0; kt < 4; ++kt) {
    int rb = kt * 16 + ((lane >> 4) << 3);
#pragma unroll
    for (int j = 0; j < 8; ++j) {
      long long o = (long long)(rb + j) * 128 + n;
      op[o] = acc[kt][j] + xp[o];
    }
  }
}

// ============================================================================
extern "C" void kernel_launch(void* const* d_in, const int* in_sizes, int n_in,
                              void* d_out, int out_size, void* d_ws, size_t ws_size,
                              hipStream_t stream) {
  const float* x  = (const float*)d_in[0];
  const float* w1 = (const float*)d_in[1];
  const float* b1 = (const float*)d_in[2];
  const float* w2 = (const float*)d_in[3];
  const float* b2 = (const float*)d_in[4];
  float* out = (float*)d_out;

  // workspace: real/imag spectrum planes, each 2*64*64*33*128 f32 (138 MB)
  float* R = (float*)d_ws;
  float* I = R + (long long)2 * 64 * 64 * 33 * 128;

  size_t smA = (size_t)(2 * 48 * 66 + 8 * 1024) * sizeof(float);               // 58112
  size_t smB = (size_t)(2 * 64 * 66 + 8 * 2048) * sizeof(float);               // 99328
  size_t smC = (size_t)(2 * 64 * 66 + 4 * 2048 + 4 * 128 + 8 * 4928) * sizeof(float); // 226304
  size_t smE = (size_t)(2 * 64 * 50 + 8 * 1536) * sizeof(float);               // 74752

  afno_rfft_w<<<8192, 256, smA, stream>>>(x, R, I);           // W: real -> 33 freqs
  afno_cdft_h<<<4224, 256, smB, stream>>>(R, I, 1.0f);        // H forward (in place)
  afno_d_mlp <<<4224, 256, smC, stream>>>(R, I, w1, b1, w2, b2); // D fwd + MLP + D inv
  afno_cdft_h<<<4224, 256, smB, stream>>>(R, I, -1.0f);       // H inverse (in place)
  afno_irfft_w<<<8192, 256, smE, stream>>>(x, R, I, out);     // W inverse + residual
}